// SKA_1812476199115
// MI455X (gfx1250) — compile-verified
//
#include <hip/hip_runtime.h>

// SKA: y[b,c,h,w] = sum_k x_pad[b,c,h+dh_k,w+dw_k] * w[b, c%16, k, h, w]
// B=8 C=128 H=W=96 WC=16 KK=9 (3x3). 8 channels share each weight plane.
// Memory-bound (AI ~1.4 FLOP/B): stream x via CDNA5 async global->LDS copies,
// reuse per-pixel weights across the 8-channel group, plain f32 FMAs.

namespace {
constexpr int B_    = 8;
constexpr int C_    = 128;
constexpr int H_    = 96;
constexpr int W_    = 96;
constexpr int WC_   = 16;
constexpr int G_    = C_ / WC_;   // 8 channel groups per weight channel
constexpr int TH_   = 8;          // tile rows per block
constexpr int TPB_  = 192;        // 24 quads/row * 8 rows
constexpr int TILES_ = H_ / TH_;  // 12
// LDS: 4-float guard | 8 buffers * (10 rows * 100-float stride) | 8-float trash
constexpr int LDSF_ = 4 + G_ * 1000 + 8;  // 8012 floats = 32048 B
}

__global__ __launch_bounds__(TPB_) void ska_kernel(const float* __restrict__ x,
                                                   const float* __restrict__ wgt,
                                                   float* __restrict__ y) {
  __shared__ __align__(16) float sh[LDSF_];

  const int tid  = threadIdx.x;
  const int blk  = blockIdx.x;
  const int tile = blk % TILES_;
  const int ci   = (blk / TILES_) % WC_;
  const int b    = blk / (TILES_ * WC_);
  const int h0   = tile * TH_;

  // ---- zero LDS: guard + pad cols + OOB rows + trash all start at 0 ----
  float4 z4; z4.x = z4.y = z4.z = z4.w = 0.f;
  float4* shv = (float4*)sh;
#pragma unroll
  for (int i = tid; i < LDSF_ / 4; i += TPB_) shv[i] = z4;
  __syncthreads();

  // ---- issue async global->LDS copies for all 8 channel groups -------
  // Each x tile: rows h0-1..h0+8 (10 rows * 96 floats), per-row b128 chunks.
  // Round 0: chunks 0..191 (rows 0..7); round 1: chunks 192..239 (rows 8,9),
  // duplicated across lanes so every wave issues exactly 2 asyncs per group.
  const unsigned shbase   = (unsigned)(size_t)&sh[0];
  const unsigned ldstrash = shbase + (4u + 8000u) * 4u;
  const int c0r = tid / 24, c0x = tid % 24;
  const int t48 = tid % 48;
  const int c1r = 8 + t48 / 24, c1x = t48 % 24;
  const int row0 = h0 - 1 + c0r;
  const int row1 = h0 - 1 + c1r;
  const bool v0 = (unsigned)row0 < (unsigned)H_;
  const bool v1 = (unsigned)row1 < (unsigned)H_;
  const unsigned l0base = shbase + (4u + (unsigned)(c0r * 100 + c0x * 4)) * 4u;
  const unsigned l1base = shbase + (4u + (unsigned)(c1r * 100 + c1x * 4)) * 4u;
  const unsigned long long xbase = (unsigned long long)x;

#pragma unroll
  for (int g = 0; g < G_; ++g) {
    const int c = g * WC_ + ci;
    const unsigned plane = (unsigned)((b * C_ + c) * (H_ * W_)) * 4u;
    // invalid boundary rows: redirect LDS dest to trash, clamp src to plane
    unsigned l0 = v0 ? (l0base + (unsigned)(g * 1000) * 4u) : ldstrash;
    unsigned o0 = v0 ? (plane + (unsigned)(row0 * W_ + c0x * 4) * 4u) : plane;
    unsigned l1 = v1 ? (l1base + (unsigned)(g * 1000) * 4u) : ldstrash;
    unsigned o1 = v1 ? (plane + (unsigned)(row1 * W_ + c1x * 4) * 4u) : plane;
    asm volatile("global_load_async_to_lds_b128 %0, %1, %2"
                 :: "v"(l0), "v"(o0), "s"(xbase) : "memory");
    asm volatile("global_load_async_to_lds_b128 %0, %1, %2"
                 :: "v"(l1), "v"(o1), "s"(xbase) : "memory");
  }

  // ---- per-pixel weights: loaded once, reused for all 8 groups -------
  const int r0 = tid / 24;
  const int w0 = (tid % 24) * 4;
  const int h  = h0 + r0;
  const float* wp = wgt + ((size_t)(b * WC_ + ci) * 9) * (H_ * W_)
                        + (size_t)h * W_ + w0;
  float4 wv[9];
#pragma unroll
  for (int k = 0; k < 9; ++k)
    wv[k] = *(const float4*)(wp + (size_t)k * (H_ * W_));
  // zero-pad taps that fall outside left/right image border
  if (w0 == 0)       { wv[0].x = 0.f; wv[3].x = 0.f; wv[6].x = 0.f; }
  if (w0 == W_ - 4)  { wv[2].w = 0.f; wv[5].w = 0.f; wv[8].w = 0.f; }

  float* yq = y + (size_t)((b * C_ + ci) * H_ + h) * W_ + w0;
  const float* xb0 = sh + 4 + r0 * 100 + w0;

  // Per group: wait for its tile (in-order async completion), barrier,
  // 3-row stencil from LDS with per-pixel weights, float4 store.
#define SKA_STEP(GG, CNT)                                                   \
  {                                                                         \
    asm volatile("s_wait_asynccnt " CNT ::: "memory");                      \
    __syncthreads();                                                        \
    const float* xb = xb0 + (GG) * 1000;                                    \
    float a0 = 0.f, a1 = 0.f, a2 = 0.f, a3 = 0.f;                           \
    _Pragma("unroll")                                                       \
    for (int dy = 0; dy < 3; ++dy) {                                        \
      const float* row = xb + dy * 100;                                     \
      float  xm = row[-1];                                                  \
      float4 xc = *(const float4*)row;                                      \
      float  xp = row[4];                                                   \
      float4 wa  = wv[dy * 3 + 0];                                          \
      float4 wb  = wv[dy * 3 + 1];                                          \
      float4 wcc = wv[dy * 3 + 2];                                          \
      a0 = fmaf(xm,   wa.x, fmaf(xc.x, wb.x, fmaf(xc.y, wcc.x, a0)));       \
      a1 = fmaf(xc.x, wa.y, fmaf(xc.y, wb.y, fmaf(xc.z, wcc.y, a1)));       \
      a2 = fmaf(xc.y, wa.z, fmaf(xc.z, wb.z, fmaf(xc.w, wcc.z, a2)));       \
      a3 = fmaf(xc.z, wa.w, fmaf(xc.w, wb.w, fmaf(xp,   wcc.w, a3)));       \
    }                                                                       \
    float4 o; o.x = a0; o.y = a1; o.z = a2; o.w = a3;                       \
    *(float4*)(yq + (size_t)((GG) * WC_) * (H_ * W_)) = o;                  \
  }

  SKA_STEP(0, "14")
  SKA_STEP(1, "12")
  SKA_STEP(2, "10")
  SKA_STEP(3, "8")
  SKA_STEP(4, "6")
  SKA_STEP(5, "4")
  SKA_STEP(6, "2")
  SKA_STEP(7, "0")
#undef SKA_STEP
}

extern "C" void kernel_launch(void* const* d_in, const int* in_sizes, int n_in,
                              void* d_out, int out_size, void* d_ws, size_t ws_size,
                              hipStream_t stream) {
  (void)in_sizes; (void)n_in; (void)out_size; (void)d_ws; (void)ws_size;
  const float* x = (const float*)d_in[0];
  const float* w = (const float*)d_in[1];
  float* y = (float*)d_out;
  dim3 grid(B_ * WC_ * TILES_);   // 8 * 16 * 12 = 1536 blocks
  dim3 block(TPB_);               // 192 threads = 6 wave32
  hipLaunchKernelGGL(ska_kernel, grid, block, 0, stream, x, w, y);
}